// Attention_23536420782780
// MI455X (gfx1250) — compile-verified
//
#include <hip/hip_runtime.h>
#include <math.h>

// Shapes (fixed by the reference)
#define BATCH     8
#define NQ        64     // N_CANDS
#define NC        512    // CTX_LEN
#define DK        512    // QUERY_SIZE == CONTEXT_SIZE
#define NH        256    // HIDDEN
#define EPSV      1e-5f

typedef __attribute__((ext_vector_type(2))) float v2f;
typedef __attribute__((ext_vector_type(8))) float v8f;

#if __has_builtin(__builtin_amdgcn_tanhf)
__device__ __forceinline__ float fast_tanh(float x) { return __builtin_amdgcn_tanhf(x); }
#else
__device__ __forceinline__ float fast_tanh(float x) { return tanhf(x); }
#endif

// ---------------------------------------------------------------------------
// Kernel 1: out[m,h] = sum_k X[m,k] * W[h,k]  (+ bias[h]), K=512, H=256.
// X: M x 512 row-major, W: 256 x 512 row-major (both K-contiguous).
// One wave32 computes a 16x64 output strip (4 accumulators) so the A fragment
// is reused 4x per K-step: 5 loads -> 4 V_WMMA_F32_16X16X4_F32.
// Fragment layouts per CDNA5 ISA 7.12.2 (fp32, wave32):
//   A  lane L, vgpr v : A[m0 + (L&15)][k0 + 2*(L>>4) + v]
//   B  lane L, vgpr v : B[k0 + 2*(L>>4) + v][n0 + (L&15)]
//   C/D lane L, vgpr v: D[m0 + v + 8*(L>>4)][n0 + (L&15)]
// ---------------------------------------------------------------------------
__global__ void gemm_xwT_bias_wmma(const float* __restrict__ X,
                                   const float* __restrict__ W,
                                   const float* __restrict__ bias,
                                   float* __restrict__ out) {
  const int wave = (blockIdx.x * blockDim.x + threadIdx.x) >> 5;
  const int lane = threadIdx.x & 31;
  const int mt = wave >> 2;             // 4 64-wide N strips cover H=256
  const int nt = wave & 3;
  const int m0 = mt << 4;
  const int h0 = nt << 6;

  const int r  = lane & 15;             // A row (M) / B col (N)
  const int kh = (lane >> 4) << 1;      // K sub-offset within a 4-step

  const float* __restrict__ Xrow = X + (size_t)(m0 + r) * DK + kh;
  const float* __restrict__ W0   = W + (size_t)(h0 + r) * DK + kh;

  v8f acc0 = {}, acc1 = {}, acc2 = {}, acc3 = {};
#pragma unroll 4
  for (int k0 = 0; k0 < DK; k0 += 4) {
    v2f a  = *(const v2f*)(Xrow + k0);
    v2f b0 = *(const v2f*)(W0 + k0);
    v2f b1 = *(const v2f*)(W0 + 16 * DK + k0);
    v2f b2 = *(const v2f*)(W0 + 32 * DK + k0);
    v2f b3 = *(const v2f*)(W0 + 48 * DK + k0);
    acc0 = __builtin_amdgcn_wmma_f32_16x16x4_f32(false, a, false, b0, (short)0, acc0, false, false);
    acc1 = __builtin_amdgcn_wmma_f32_16x16x4_f32(false, a, false, b1, (short)0, acc1, false, false);
    acc2 = __builtin_amdgcn_wmma_f32_16x16x4_f32(false, a, false, b2, (short)0, acc2, false, false);
    acc3 = __builtin_amdgcn_wmma_f32_16x16x4_f32(false, a, false, b3, (short)0, acc3, false, false);
  }

  const int mb = (lane >> 4) << 3;      // 0 or 8
  float bs0 = 0.f, bs1 = 0.f, bs2 = 0.f, bs3 = 0.f;
  if (bias) {
    bs0 = bias[h0 + r];
    bs1 = bias[h0 + 16 + r];
    bs2 = bias[h0 + 32 + r];
    bs3 = bias[h0 + 48 + r];
  }
  float* __restrict__ orow = out + (size_t)(m0 + mb) * NH + h0 + r;
#pragma unroll
  for (int v = 0; v < 8; ++v) {
    orow[(size_t)v * NH]      = acc0[v] + bs0;
    orow[(size_t)v * NH + 16] = acc1[v] + bs1;
    orow[(size_t)v * NH + 32] = acc2[v] + bs2;
    orow[(size_t)v * NH + 48] = acc3[v] + bs3;
  }
}

// ---------------------------------------------------------------------------
// Kernel 2: fused logit + masked softmax. One 256-thread block per (b,q).
// Each thread owns contexts c = t and c = t + 256. LDS stages the res_q row,
// W_o, and the reduction tree. Hardware V_TANH_F32 / V_EXP_F32 when available.
//   logit = sum_h W_o[h] * tanh(res_c[b,c,h] + res_q[b,q,h]) + b_o
//   weights = mask*exp(logit) / (sum_c mask*exp(logit) + eps)
// ---------------------------------------------------------------------------
__global__ void logits_softmax(const float* __restrict__ resc,
                               const float* __restrict__ resq,
                               const float* __restrict__ mask,
                               const float* __restrict__ Wo,
                               const float* __restrict__ b_o,
                               float* __restrict__ weights) {
  const int bq = blockIdx.x;            // 0 .. B*NQ-1
  const int b  = bq >> 6;               // / NQ
  const int t  = threadIdx.x;           // 0..255

  __shared__ float s_rq[NH];
  __shared__ float s_wo[NH];
  __shared__ float s_red[256];

  s_rq[t] = resq[(size_t)bq * NH + t];
  s_wo[t] = Wo[t];
  __syncthreads();

  const float bo = b_o[0];
  float wexp0 = 0.0f, wexp1 = 0.0f, local = 0.0f;

#pragma unroll
  for (int i = 0; i < 2; ++i) {
    const int c = t + (i << 8);
    const float4* __restrict__ rc =
        (const float4*)(resc + ((size_t)(b * NC + c)) * NH);
    float acc = 0.0f;
#pragma unroll 4
    for (int h4 = 0; h4 < NH / 4; ++h4) {
      float4 v = rc[h4];
      const int h = h4 << 2;
      acc += s_wo[h + 0] * fast_tanh(v.x + s_rq[h + 0]);
      acc += s_wo[h + 1] * fast_tanh(v.y + s_rq[h + 1]);
      acc += s_wo[h + 2] * fast_tanh(v.z + s_rq[h + 2]);
      acc += s_wo[h + 3] * fast_tanh(v.w + s_rq[h + 3]);
    }
    const float e = mask[(size_t)b * NC + c] * __expf(acc + bo);
    if (i == 0) wexp0 = e; else wexp1 = e;
    local += e;
  }

  s_red[t] = local;
  __syncthreads();
#pragma unroll
  for (int s = 128; s > 0; s >>= 1) {
    if (t < s) s_red[t] += s_red[t + s];
    __syncthreads();
  }
  const float inv = 1.0f / (s_red[0] + EPSV);

  weights[(size_t)bq * NC + t]       = wexp0 * inv;
  weights[(size_t)bq * NC + t + 256] = wexp1 * inv;
}

// ---------------------------------------------------------------------------
// Kernel 3: output[b,q,d] = sum_c weights[b,q,c] * context[b,c,d]
// Per batch: (64 x 512) @ (512 x 512). One wave per 16x32 strip (2 accs);
// A (weights) is K-contiguous (v2f loads); B (context) is N-contiguous,
// so each lane does two stride-DK b32 loads per tile per k-step.
// ---------------------------------------------------------------------------
__global__ void out_gemm_wmma(const float* __restrict__ weights,
                              const float* __restrict__ context,
                              float* __restrict__ out) {
  const int wave = (blockIdx.x * blockDim.x + threadIdx.x) >> 5;
  const int lane = threadIdx.x & 31;
  const int b    = wave >> 6;           // 64 strips per batch
  const int tile = wave & 63;
  const int q0   = (tile >> 4) << 4;    // 4 q-tiles
  const int d0   = (tile & 15) << 5;    // 16 32-wide d-strips

  const int r  = lane & 15;
  const int kh = (lane >> 4) << 1;

  const float* __restrict__ Wrow = weights + (size_t)(b * NQ + q0 + r) * NC + kh;
  const float* __restrict__ Ctx  = context + (size_t)b * NC * DK + d0 + r;

  v8f acc0 = {}, acc1 = {};
#pragma unroll 4
  for (int k0 = 0; k0 < NC; k0 += 4) {
    v2f a = *(const v2f*)(Wrow + k0);
    const float* c0 = Ctx + (size_t)(k0 + kh) * DK;
    v2f b0, b1;
    b0.x = c0[0];       b0.y = c0[DK];
    b1.x = c0[16];      b1.y = c0[DK + 16];
    acc0 = __builtin_amdgcn_wmma_f32_16x16x4_f32(false, a, false, b0, (short)0, acc0, false, false);
    acc1 = __builtin_amdgcn_wmma_f32_16x16x4_f32(false, a, false, b1, (short)0, acc1, false, false);
  }

  const int mb = (lane >> 4) << 3;
  float* __restrict__ orow = out + (size_t)(b * NQ + q0 + mb) * DK + d0 + r;
#pragma unroll
  for (int v = 0; v < 8; ++v) {
    orow[(size_t)v * DK]      = acc0[v];
    orow[(size_t)v * DK + 16] = acc1[v];
  }
}

// ---------------------------------------------------------------------------
extern "C" void kernel_launch(void* const* d_in, const int* in_sizes, int n_in,
                              void* d_out, int out_size, void* d_ws, size_t ws_size,
                              hipStream_t stream) {
  const float* query   = (const float*)d_in[0];   // (B, NQ, DK)
  const float* context = (const float*)d_in[1];   // (B, NC, DK)
  const float* mask    = (const float*)d_in[2];   // (B, NC)
  const float* W_c     = (const float*)d_in[3];   // (NH, DK)
  const float* b_c     = (const float*)d_in[4];   // (NH,)
  const float* W_q     = (const float*)d_in[5];   // (NH, DK)
  const float* W_o     = (const float*)d_in[6];   // (NH,)
  const float* b_o     = (const float*)d_in[7];   // scalar

  float* out     = (float*)d_out;                 // (B, NQ, DK)
  float* weights = out + (size_t)BATCH * NQ * DK; // (B, NQ, NC)

  float* resc = (float*)d_ws;                     // (B*NC, NH) = 4 MB
  float* resq = resc + (size_t)BATCH * NC * NH;   // (B*NQ, NH) = 512 KB

  // res_c = context @ W_c^T + b_c : M = B*NC = 4096 -> 256 Mtiles x 4 strips
  //   = 1024 waves = 128 blocks of 8 waves
  gemm_xwT_bias_wmma<<<128, 256, 0, stream>>>(context, W_c, b_c, resc);
  // res_q = query @ W_q^T : M = B*NQ = 512 -> 32 Mtiles x 4 strips
  //   = 128 waves = 16 blocks
  gemm_xwT_bias_wmma<<<16, 256, 0, stream>>>(query, W_q, nullptr, resq);
  // fused tanh-logit + masked softmax: one block per (b,q)
  logits_softmax<<<BATCH * NQ, 256, 0, stream>>>(resc, resq, mask, W_o, b_o,
                                                 weights);
  // output = weights @ context : 8 batches * 64 strips = 512 waves = 64 blocks
  out_gemm_wmma<<<64, 256, 0, stream>>>(weights, context, out);
}